// LGSSM_65231963291685
// MI455X (gfx1250) — compile-verified
//
#include <hip/hip_runtime.h>
#include <hip/hip_bf16.h>
#include <math.h>

typedef float v2f __attribute__((ext_vector_type(2)));
typedef float v8f __attribute__((ext_vector_type(8)));

namespace {
constexpr int Dd  = 128;   // state dim
constexpr int Nn  = 64;    // obs dim
constexpr int Tt  = 4096;  // time steps
constexpr int SLD = 65;    // padded leading dim for S (avoid LDS bank conflicts)
}

// ---- WMMA f32 16x16x4: D = A(16x4) * B(4x16) + C(16x16) --------------------
__device__ __forceinline__ v8f wmma4(v2f a, v2f b, v8f c) {
  return __builtin_amdgcn_wmma_f32_16x16x4_f32(false, a, false, b, (short)0, c,
                                               false, false);
}

// A-fragment (16 rows x 4 K): lanes0-15 K={k0,k0+1}, lanes16-31 K={k0+2,k0+3}
__device__ __forceinline__ v2f afrag(const float* s, int ld, int r0, int k0,
                                     int l16, int half) {
  const float* p = s + (r0 + l16) * ld + k0 + 2 * half;
  v2f a; a.x = p[0]; a.y = p[1];
  return a;
}
// A-fragment with logical A[r][k] = s[k*ld + r] (transposed source)
__device__ __forceinline__ v2f afragT(const float* s, int ld, int r0, int k0,
                                      int l16, int half) {
  int r = r0 + l16, k = k0 + 2 * half;
  v2f a; a.x = s[k * ld + r]; a.y = s[(k + 1) * ld + r];
  return a;
}
// B-fragment (4 K x 16 cols) from row-major src
__device__ __forceinline__ v2f bfrag(const float* s, int ld, int k0, int c0,
                                     int l16, int half) {
  int k = k0 + 2 * half, c = c0 + l16;
  v2f b; b.x = s[k * ld + c]; b.y = s[(k + 1) * ld + c];
  return b;
}
// B-fragment with logical B[k][c] = s[c*ld + k] (transposed source)
__device__ __forceinline__ v2f bfragT(const float* s, int ld, int k0, int c0,
                                      int l16, int half) {
  int k = k0 + 2 * half, c = c0 + l16;
  v2f b; b.x = s[c * ld + k]; b.y = s[c * ld + k + 1];
  return b;
}

__global__ __launch_bounds__(256) void lgssm_kf_kernel(
    const float* __restrict__ gy,    // [T,N]
    const float* __restrict__ gA,    // [D,D]
    const float* __restrict__ gC,    // [N,D]
    const float* __restrict__ gQ,    // [D,D]
    const float* __restrict__ gR,    // [N,N]
    const float* __restrict__ gx0mu, // [D]
    const float* __restrict__ gx0s,  // [D,D]
    float* __restrict__ out) {
  // LDS: 3*64KB + 2*32KB + ~17KB ≈ 281 KB of the WGP's 320 KB
  __shared__ float sA[Dd * Dd];   // A
  __shared__ float sP[Dd * Dd];   // P -> P_minus -> P_new
  __shared__ float sT[Dd * Dd];   // temp: A@P, later X = S^{-1} CP (rows 0..63)
  __shared__ float sC[Nn * Dd];   // C
  __shared__ float sCP[Nn * Dd];  // C @ P_minus
  __shared__ float sS[Nn * SLD];  // innovation covariance / Cholesky factor
  __shared__ float sLd[Nn];       // Cholesky diagonal
  __shared__ float sm[Dd], smm[Dd], sdy[Nn], sz[Nn];
  __shared__ float sAccum;

  const int tid  = threadIdx.x;
  const int wave = tid >> 5;   // 0..7 (wave32)
  const int lane = tid & 31;
  const int half = lane >> 4;
  const int l16  = lane & 15;

  const float CONST_Z = -0.5f * (float)Nn * 1.83787706640934534f;

  for (int i = tid; i < Dd * Dd; i += 256) { sA[i] = gA[i]; sP[i] = gx0s[i]; }
  for (int i = tid; i < Nn * Dd; i += 256) sC[i] = gC[i];
  if (tid < Dd) sm[tid] = gx0mu[tid];
  if (tid == 0) sAccum = 0.0f;
  __syncthreads();

  float* outAccum = out;                    // [1]
  float* outLogp  = out + 1;                // [T]
  float* outMu    = out + 1 + Tt;           // [T,D]
  float* outSig   = out + 1 + Tt + Tt * Dd; // [T,D,D]

#pragma unroll 1
  for (int t = 0; t < Tt; ++t) {
    if (tid == 0 && t + 1 < Tt)
      __builtin_prefetch(&gy[(t + 1) * Nn], 0, 1);  // global_prefetch_b8

    // ---- Phase 1: T = A @ P : each wave owns one 16-row strip x 128 cols --
    {
      const int r0 = wave << 4;
      v8f acc[8];
#pragma unroll
      for (int c = 0; c < 8; ++c) acc[c] = (v8f){};
#pragma unroll 2
      for (int k0 = 0; k0 < Dd; k0 += 4) {
        v2f a = afrag(sA, Dd, r0, k0, l16, half);  // 1 A-frag reused 8x
#pragma unroll
        for (int c = 0; c < 8; ++c)
          acc[c] = wmma4(a, bfrag(sP, Dd, k0, c << 4, l16, half), acc[c]);
      }
#pragma unroll
      for (int c = 0; c < 8; ++c)
#pragma unroll
        for (int i = 0; i < 8; ++i)
          sT[(r0 + 8 * half + i) * Dd + (c << 4) + l16] = acc[c][i];
    }
    __syncthreads();

    // ---- Phase 2: P_minus = T @ A^T + Q ----------------------------------
    {
      const int r0 = wave << 4;
      v8f acc[8];
      const float* qp = gQ + (r0 + 8 * half) * Dd + l16;  // base + IOFFSET
#pragma unroll
      for (int c = 0; c < 8; ++c)
#pragma unroll
        for (int i = 0; i < 8; ++i) acc[c][i] = qp[i * Dd + (c << 4)];
#pragma unroll 2
      for (int k0 = 0; k0 < Dd; k0 += 4) {
        v2f a = afrag(sT, Dd, r0, k0, l16, half);
#pragma unroll
        for (int c = 0; c < 8; ++c)
          acc[c] = wmma4(a, bfragT(sA, Dd, k0, c << 4, l16, half), acc[c]);
      }
#pragma unroll
      for (int c = 0; c < 8; ++c)
#pragma unroll
        for (int i = 0; i < 8; ++i)
          sP[(r0 + 8 * half + i) * Dd + (c << 4) + l16] = acc[c][i];
    }
    __syncthreads();

    // ---- Phase 3: CP = C @ P_minus; m_minus = A @ m -----------------------
    // 4 row strips of CP; 2 waves per strip, 64 cols each.
    {
      const int r0 = (wave >> 1) << 4;
      const int cb = (wave & 1) << 6;
      v8f acc[4];
#pragma unroll
      for (int c = 0; c < 4; ++c) acc[c] = (v8f){};
#pragma unroll 2
      for (int k0 = 0; k0 < Dd; k0 += 4) {
        v2f a = afrag(sC, Dd, r0, k0, l16, half);
#pragma unroll
        for (int c = 0; c < 4; ++c)
          acc[c] = wmma4(a, bfrag(sP, Dd, k0, cb + (c << 4), l16, half), acc[c]);
      }
#pragma unroll
      for (int c = 0; c < 4; ++c)
#pragma unroll
        for (int i = 0; i < 8; ++i)
          sCP[(r0 + 8 * half + i) * Dd + cb + (c << 4) + l16] = acc[c][i];
    }
    if (tid < Dd) {
      float acc = 0.0f;
      for (int j = 0; j < Dd; ++j) acc += sA[tid * Dd + j] * sm[j];
      smm[tid] = acc;
    }
    __syncthreads();

    // ---- Phase 4: S = CP @ C^T + R; dy = y_t - C m_minus ------------------
    // 4 row strips x 4 col tiles; wave w: strip w>>1, col pair (w&1)*2.
    {
      const int r0 = (wave >> 1) << 4;
      const int cb = (wave & 1) << 5;
      v8f acc[2];
      const float* rp = gR + (r0 + 8 * half) * Nn + cb + l16;
#pragma unroll
      for (int c = 0; c < 2; ++c)
#pragma unroll
        for (int i = 0; i < 8; ++i) acc[c][i] = rp[i * Nn + (c << 4)];
#pragma unroll 2
      for (int k0 = 0; k0 < Dd; k0 += 4) {
        v2f a = afrag(sCP, Dd, r0, k0, l16, half);
#pragma unroll
        for (int c = 0; c < 2; ++c)
          acc[c] = wmma4(a, bfragT(sC, Dd, k0, cb + (c << 4), l16, half), acc[c]);
      }
#pragma unroll
      for (int c = 0; c < 2; ++c)
#pragma unroll
        for (int i = 0; i < 8; ++i)
          sS[(r0 + 8 * half + i) * SLD + cb + (c << 4) + l16] = acc[c][i];
    }
    if (tid < Nn) {
      float acc = gy[t * Nn + tid];
      for (int j = 0; j < Dd; ++j) acc -= sC[tid * Dd + j] * smm[j];
      sdy[tid] = acc;
    }
    __syncthreads();

    // symmetrize S = 0.5*(S + S^T): one thread per unordered pair
    for (int i = tid; i < Nn * Nn; i += 256) {
      int r = i >> 6, c = i & 63;
      if (r < c) {
        float v = 0.5f * (sS[r * SLD + c] + sS[c * SLD + r]);
        sS[r * SLD + c] = v;
        sS[c * SLD + r] = v;
      }
    }
    __syncthreads();

    // ---- Phase 5: Cholesky of S (lower), diag kept in sLd -----------------
#pragma unroll 1
    for (int j = 0; j < Nn; ++j) {
      float d = sqrtf(sS[j * SLD + j]);
      if (tid == 0) sLd[j] = d;
      if (tid > j && tid < Nn) sS[tid * SLD + j] /= d;
      __syncthreads();
      if (tid > j && tid < Nn) {
        float lij = sS[tid * SLD + j];
        for (int k = j + 1; k <= tid; ++k)
          sS[tid * SLD + k] -= lij * sS[k * SLD + j];
      }
      __syncthreads();
    }

    // ---- Phase 6: X = S^{-1} CP via two triangular solves (into sT) -------
    if (tid < Dd) {
      const int c = tid;
      for (int i = 0; i < Nn; ++i) {            // forward: L W = CP
        float w = sCP[i * Dd + c];
        for (int k = 0; k < i; ++k) w -= sS[i * SLD + k] * sT[k * Dd + c];
        sT[i * Dd + c] = w / sLd[i];
      }
      for (int i = Nn - 1; i >= 0; --i) {       // backward: L^T X = W
        float w = sT[i * Dd + c];
        for (int k = i + 1; k < Nn; ++k) w -= sS[k * SLD + i] * sT[k * Dd + c];
        sT[i * Dd + c] = w / sLd[i];
      }
    }
    if (tid == 255) {                           // z = L^{-1} dy, logp
      float zz = 0.0f, sld = 0.0f;
      for (int i = 0; i < Nn; ++i) {
        float w = sdy[i];
        for (int k = 0; k < i; ++k) w -= sS[i * SLD + k] * sz[k];
        w /= sLd[i];
        sz[i] = w;
        zz += w * w;
        sld += logf(sLd[i]);
      }
      float logp = -0.5f * zz - sld + CONST_Z;
      outLogp[t] = logp;
      sAccum += logp;
    }
    __syncthreads();

    // ---- Phase 7: m_new = m_minus + X^T dy --------------------------------
    if (tid < Dd) {
      float acc = smm[tid];
      for (int i = 0; i < Nn; ++i) acc += sT[i * Dd + tid] * sdy[i];
      sm[tid] = acc;
      outMu[(size_t)t * Dd + tid] = acc;
    }

    // ---- Phase 8: P_new = P_minus - CP^T @ X (128x128x64) -----------------
    {
      const int r0 = wave << 4;
      v8f acc[8];
#pragma unroll
      for (int c = 0; c < 8; ++c) acc[c] = (v8f){};
#pragma unroll 2
      for (int k0 = 0; k0 < Nn; k0 += 4) {
        v2f a = afragT(sCP, Dd, r0, k0, l16, half);  // A[r][k] = CP[k][r]
#pragma unroll
        for (int c = 0; c < 8; ++c)
          acc[c] = wmma4(a, bfrag(sT, Dd, k0, c << 4, l16, half), acc[c]);
      }
#pragma unroll
      for (int c = 0; c < 8; ++c)
#pragma unroll
        for (int i = 0; i < 8; ++i) {
          int r = r0 + 8 * half + i, cc = (c << 4) + l16;
          float p = sP[r * Dd + cc] - acc[c][i];
          sP[r * Dd + cc] = p;
          outSig[(size_t)t * Dd * Dd + r * Dd + cc] = p;
        }
    }
    __syncthreads();
  }

  if (tid == 255) outAccum[0] = sAccum;
}

extern "C" void kernel_launch(void* const* d_in, const int* in_sizes, int n_in,
                              void* d_out, int out_size, void* d_ws,
                              size_t ws_size, hipStream_t stream) {
  const float* y    = (const float*)d_in[0];
  const float* A    = (const float*)d_in[1];
  const float* C    = (const float*)d_in[2];
  const float* Q    = (const float*)d_in[3];
  const float* R    = (const float*)d_in[4];
  const float* x0mu = (const float*)d_in[5];
  const float* x0s  = (const float*)d_in[6];
  float* out = (float*)d_out;
  (void)in_sizes; (void)n_in; (void)out_size; (void)d_ws; (void)ws_size;

  // Sequential scan: one resident workgroup (8 wave32s on one WGP) owns the
  // whole filter; state lives in LDS across all 4096 steps.
  lgssm_kf_kernel<<<1, 256, 0, stream>>>(y, A, C, Q, R, x0mu, x0s, out);
}